// Attention_19473381720489
// MI455X (gfx1250) — compile-verified
//
#include <hip/hip_runtime.h>

typedef __attribute__((ext_vector_type(2))) float    v2f;
typedef __attribute__((ext_vector_type(4))) float    v4f;
typedef __attribute__((ext_vector_type(8))) float    v8f;
typedef __attribute__((ext_vector_type(4))) unsigned v4u;
typedef __attribute__((ext_vector_type(4))) int      v4i;
typedef __attribute__((ext_vector_type(8))) int      v8i;

#define IN_DIM 256
#define HIDDEN 16
#define NSEG 512
#define ROWS_PER_BLOCK 128
#define KCHUNK 64
#define NCHUNKS (IN_DIM / KCHUNK)
#define XPAD 68   // floats per LDS row of x tile (TDM pad: 64 dw + 4 dw pad)
#define HPAD 18   // floats per LDS row of H tile (72B: 8B aligned, conflict-free)

// Order-preserving float -> uint key (for atomicMax-based float max)
__device__ __forceinline__ unsigned fkey(float f) {
    unsigned u = __float_as_uint(f);
    return (u & 0x80000000u) ? ~u : (u | 0x80000000u);
}
__device__ __forceinline__ float funkey(unsigned k) {
    unsigned u = (k & 0x80000000u) ? (k & 0x7fffffffu) : ~k;
    return __uint_as_float(u);
}

// Issue one TDM 2D tile load: 128 rows x 64 f32 cols, row stride 256 f32 in
// memory, zero-fill rows >= rowsLeft, LDS rows padded 64->68 dwords.
// Descriptor per CDNA5 ISA §8.3/8.4 (group0 128b, group1 256b, groups 2/3 zero).
// This toolchain exposes the 6-arg builtin: (v4u, v8i, v4i, v4i, v8i, i32 cpol).
__device__ __forceinline__ void tdm_load_chunk(const float* gptr, unsigned lds_off,
                                               int rowsLeft)
{
    unsigned long long ga = (unsigned long long)(uintptr_t)gptr;
    v4u g0;
    g0.x = 1u;                                            // count=1, user mode
    g0.y = lds_off;                                       // lds_addr [63:32]
    g0.z = (unsigned)(ga & 0xffffffffu);                  // global_addr lo
    g0.w = (unsigned)((ga >> 32) & 0x01ffffffu)           // global_addr hi (57b)
         | (2u << 30);                                    // type=2 ("image")
    v8i g1;
    g1[0] = (2 << 16)        // data_size = 4B
          | (1 << 20)        // pad_enable
          | (5 << 22)        // pad_interval: 2^(5+1) = 64 dwords
          | (3 << 25);       // pad_amount: 3+1 = 4 dwords  -> stride 68 dw
    g1[1] = (KCHUNK & 0xffff) << 16;                      // tensor_dim0 lo16 = 64
    g1[2] = ((rowsLeft & 0xffff) << 16);                  // dim0 hi16=0 | tensor_dim1 lo16
    g1[3] = ((rowsLeft >> 16) & 0xffff)                   // tensor_dim1 hi16
          | (KCHUNK << 16);                               // tile_dim0 = 64
    g1[4] = ROWS_PER_BLOCK;                               // tile_dim1 = 128 (tile_dim2=0)
    g1[5] = IN_DIM;                                       // tensor_dim0_stride = 256
    g1[6] = 0;                                            // stride hi / dim1_stride lo
    g1[7] = 0;
    v4i z4 = {0, 0, 0, 0};                                // groups 2/3: 2D tensor
    v8i z8 = {0, 0, 0, 0, 0, 0, 0, 0};
    __builtin_amdgcn_tensor_load_to_lds(g0, g1, z4, z4, z8, 0);
}

__global__ __launch_bounds__(256) void mlp_pool_kernel(
    const float* __restrict__ x, const int* __restrict__ batch,
    const float* __restrict__ w1, const float* __restrict__ b1,
    const float* __restrict__ w2, const float* __restrict__ b2,
    unsigned* __restrict__ mxkey, float* __restrict__ sm, int nrows)
{
    __shared__ __align__(16) float xs[2][ROWS_PER_BLOCK * XPAD]; // 2 x 34816 B
    __shared__ __align__(16) float w1s[IN_DIM * HIDDEN];         // 16384 B
    __shared__ __align__(16) float w2s[HIDDEN * IN_DIM];         // 16384 B
    __shared__ __align__(16) float hs[8 * 16 * HPAD];            //  9216 B
    __shared__ float b1s[HIDDEN];
    __shared__ float b2s[IN_DIM];
    __shared__ int   batch_s[ROWS_PER_BLOCK];

    const int tid  = threadIdx.x;
    const int lane = tid & 31;
    const int wv   = tid >> 5;
    const int lmod = lane & 15;
    const int lhi  = lane >> 4;            // 0: rows 0-7 / K+{0,1} ; 1: rows 8-15 / K+{2,3}
    const int rowBase = blockIdx.x * ROWS_PER_BLOCK;
    const int rowsLeft = nrows - rowBase;  // >= 1 by grid sizing
    const int waveRow = wv * 16;

    // Cache weights / biases / batch slice in LDS
    for (int i = tid; i < IN_DIM * HIDDEN; i += 256) w1s[i] = w1[i];
    for (int i = tid; i < HIDDEN * IN_DIM; i += 256) w2s[i] = w2[i];
    if (tid < HIDDEN) b1s[tid] = b1[tid];
    if (tid < IN_DIM) b2s[tid] = b2[tid];
    if (tid < ROWS_PER_BLOCK) {
        int g = rowBase + tid;
        batch_s[tid] = batch[g < nrows ? g : (nrows - 1)];
    }

    // Kick off the first TDM chunk (wave 0 issues; EXEC is ignored by TDM)
    const float* xblk = x + (size_t)rowBase * IN_DIM;
    if (wv == 0) {
        tdm_load_chunk(xblk, (unsigned)(uintptr_t)&xs[0][0], rowsLeft);
    }
    __syncthreads();   // weights/batch visible to all waves

    // ---- Layer 1: acc1 = X_tile(16x256) @ W1(256x16) + b1, 64 x wmma K=4 ----
    v8f acc1;
    {
        float b1v = b1s[lmod];             // C/D layout: N = lane%16
        #pragma unroll
        for (int v = 0; v < 8; ++v) acc1[v] = b1v;
    }

    for (int c = 0; c < NCHUNKS; ++c) {
        if (wv == 0) __builtin_amdgcn_s_wait_tensorcnt((short)0);  // chunk c in LDS
        __syncthreads();                    // publish buffer c&1 to all waves
        if (wv == 0 && c + 1 < NCHUNKS) {   // prefetch next chunk into other buffer
            tdm_load_chunk(xblk + (c + 1) * KCHUNK,
                           (unsigned)(uintptr_t)&xs[(c + 1) & 1][0], rowsLeft);
        }

        const float* xb = &xs[c & 1][0];
        const int k0 = c * KCHUNK;
        #pragma unroll
        for (int kk = 0; kk < KCHUNK / 4; ++kk) {
            const int koff = kk * 4 + lhi * 2;
            // A frag: row (lane%16) of this wave's tile, K pair {koff, koff+1}
            v2f a = *(const v2f*)(xb + (waveRow + lmod) * XPAD + koff);
            // B frag: w1 rows {k, k+1} ({k+2,k+3} for lanes 16-31), col lane%16
            const int bk = k0 + koff;
            v2f b;
            b.x = w1s[bk * HIDDEN + lmod];
            b.y = w1s[(bk + 1) * HIDDEN + lmod];
            acc1 = __builtin_amdgcn_wmma_f32_16x16x4_f32(
                false, a, false, b, (short)0, acc1, false, false);
        }
        __syncthreads();                    // all reads of buffer c&1 done
    }

    // ReLU, then stage H (D layout -> row-major LDS) for layer-2 A frags
    float* hw = hs + wv * 16 * HPAD;
    #pragma unroll
    for (int v = 0; v < 8; ++v) {
        int m = v + lhi * 8;
        hw[m * HPAD + lmod] = fmaxf(acc1[v], 0.0f);
    }
    // wave-private LDS region; same-wave DS ops complete in order

    // ---- Layer 2 + fused segment pooling, one 16x16 N-tile at a time ----
    for (int nt = 0; nt < IN_DIM / 16; ++nt) {
        v8f acc2;
        {
            float b2v = b2s[nt * 16 + lmod];
            #pragma unroll
            for (int v = 0; v < 8; ++v) acc2[v] = b2v;
        }
        #pragma unroll
        for (int kk = 0; kk < HIDDEN / 4; ++kk) {
            const int koff = kk * 4 + lhi * 2;
            v2f a = *(const v2f*)(hw + lmod * HPAD + koff);
            v2f b;
            b.x = w2s[koff * IN_DIM + nt * 16 + lmod];
            b.y = w2s[(koff + 1) * IN_DIM + nt * 16 + lmod];
            acc2 = __builtin_amdgcn_wmma_f32_16x16x4_f32(
                false, a, false, b, (short)0, acc2, false, false);
        }

        // Run-compressed segment max/sum over this lane's 8 rows (batch sorted)
        float q[8];
        #pragma unroll
        for (int v = 0; v < 8; ++v) q[v] = acc2[v];

        const int col = nt * 16 + lmod;
        int curseg = -1;
        float rmax = 0.0f, rsum = 0.0f;
        #pragma unroll
        for (int v = 0; v < 8; ++v) {
            int m = v + lhi * 8;
            int g = rowBase + waveRow + m;
            if (g < nrows) {
                int   seg = batch_s[waveRow + m];
                float val = q[v];
                if (seg != curseg) {
                    if (curseg >= 0) {
                        atomicMax(&mxkey[(size_t)curseg * IN_DIM + col], fkey(rmax));
                        atomicAdd(&sm[(size_t)curseg * IN_DIM + col], rsum);
                    }
                    curseg = seg; rmax = val; rsum = val;
                } else {
                    rmax = fmaxf(rmax, val); rsum += val;
                }
            }
        }
        if (curseg >= 0) {
            atomicMax(&mxkey[(size_t)curseg * IN_DIM + col], fkey(rmax));
            atomicAdd(&sm[(size_t)curseg * IN_DIM + col], rsum);
        }
    }
}

__global__ __launch_bounds__(IN_DIM) void gate_kernel(
    const unsigned* __restrict__ mxkey, const float* __restrict__ sm,
    const int* __restrict__ batch, int nrows, float* __restrict__ gate)
{
    const int s = blockIdx.x;
    __shared__ int cnt_s;
    if (threadIdx.x == 0) {
        int lo = 0, hi = nrows;            // first idx with batch[i] >= s
        while (lo < hi) { int mid = (lo + hi) >> 1; if (batch[mid] < s) lo = mid + 1; else hi = mid; }
        int lo2 = lo, hi2 = nrows;         // first idx with batch[i] > s
        while (lo2 < hi2) { int mid = (lo2 + hi2) >> 1; if (batch[mid] <= s) lo2 = mid + 1; else hi2 = mid; }
        cnt_s = lo2 - lo;
    }
    __syncthreads();
    const int cnt = cnt_s;
    const int idx = s * IN_DIM + threadIdx.x;
    float g = 0.0f;
    if (cnt > 0) {
        float mx   = funkey(mxkey[idx]);
        float mean = sm[idx] / (float)cnt;
        float z    = mx + mean;
        g = 1.0f / (1.0f + expf(-z));
    }
    gate[idx] = g;
}

__global__ __launch_bounds__(256) void modulate_kernel(
    const float* __restrict__ x, const int* __restrict__ batch,
    const float* __restrict__ gate, float* __restrict__ out, int nrows)
{
    long long idx4  = (long long)blockIdx.x * 256 + threadIdx.x;
    long long total = (long long)nrows * (IN_DIM / 4);
    if (idx4 >= total) return;
    int row = (int)(idx4 >> 6);            // 64 float4 per row
    int c4  = ((int)idx4 & 63) * 4;
    int s   = batch[row];
    v4f xv = *(const v4f*)(x    + (size_t)row * IN_DIM + c4);
    v4f gv = *(const v4f*)(gate + (size_t)s   * IN_DIM + c4);
    *(v4f*)(out + (size_t)row * IN_DIM + c4) = xv * gv;
}

extern "C" void kernel_launch(void* const* d_in, const int* in_sizes, int n_in,
                              void* d_out, int out_size, void* d_ws, size_t ws_size,
                              hipStream_t stream) {
    const float* x  = (const float*)d_in[0];
    const int* batch = (const int*)d_in[1];
    const float* w1 = (const float*)d_in[2];
    const float* b1 = (const float*)d_in[3];
    const float* w2 = (const float*)d_in[4];
    const float* b2 = (const float*)d_in[5];
    const int nrows = in_sizes[1];
    float* out = (float*)d_out;

    unsigned* mxkey = (unsigned*)d_ws;
    float* sm   = (float*)((char*)d_ws + (size_t)NSEG * IN_DIM * 4);
    float* gate = (float*)((char*)d_ws + (size_t)2 * NSEG * IN_DIM * 4);

    // zero mxkey (uint-key identity) and sm each call (graph-capture safe)
    (void)hipMemsetAsync(d_ws, 0, (size_t)2 * NSEG * IN_DIM * 4, stream);

    int nb1 = (nrows + ROWS_PER_BLOCK - 1) / ROWS_PER_BLOCK;
    mlp_pool_kernel<<<nb1, 256, 0, stream>>>(x, batch, w1, b1, w2, b2, mxkey, sm, nrows);

    gate_kernel<<<NSEG, IN_DIM, 0, stream>>>(mxkey, sm, batch, nrows, gate);

    long long total4 = (long long)nrows * (IN_DIM / 4);
    int nb3 = (int)((total4 + 255) / 256);
    modulate_kernel<<<nb3, 256, 0, stream>>>(x, batch, gate, out, nrows);
}